// FutureFutureCrossAttention_89515708383692
// MI455X (gfx1250) — compile-verified
//
#include <hip/hip_runtime.h>
#include <hip/hip_bf16.h>

typedef __attribute__((ext_vector_type(2))) float v2f;
typedef __attribute__((ext_vector_type(8))) float v8f;

// Problem constants (from reference setup_inputs)
constexpr int kB  = 8;
constexpr int kA  = 64;
constexpr int kK  = 16;    // neighbours
constexpr int kL  = 30;    // sequence length
constexpr int kD  = 128;   // d_model in
constexpr int kM  = 128;   // d_model out
constexpr int kH  = 8;     // heads
constexpr int kHD = 16;    // head dim
constexpr int kR  = kB * kA * kL;        // 15360 rows for projections
constexpr int kCL = kK * kL;             // 480 key columns
constexpr float kSCALE = 0.25f;          // 16^-0.5

__device__ __forceinline__ v8f wmma_f32x4(v2f a, v2f b, v8f c) {
  // D = A(16x4) * B(4x16) + C, fp32 everywhere
  return __builtin_amdgcn_wmma_f32_16x16x4_f32(
      /*neg_a=*/false, a, /*neg_b=*/false, b,
      /*c_mod=*/(short)0, c, /*reuse_a=*/false, /*reuse_b=*/false);
}

// ---------------------------------------------------------------------------
// Kernel 1: fused Q/K/V projection.  X[15360,128] @ {Wq,Wk,Wv}[128,128] + bias
// One WG = 16 rows of X; wave w owns output columns [16w,16w+16) of all three
// outputs; the A fragment (X strip in LDS) is shared across the 3 WMMAs.
// ---------------------------------------------------------------------------
__global__ __launch_bounds__(256) void qkv_proj_kernel(
    const float* __restrict__ X,
    const float* __restrict__ Wq, const float* __restrict__ bq,
    const float* __restrict__ Wk, const float* __restrict__ bk,
    const float* __restrict__ Wv, const float* __restrict__ bv,
    float* __restrict__ qo, float* __restrict__ ko, float* __restrict__ vo) {
  constexpr int XS = 132;                 // padded LDS row stride (bank-safe)
  __shared__ float Xs[16 * XS];
  const int tid   = threadIdx.x;
  const int wave  = tid >> 5;
  const int lane  = tid & 31;
  const int lhalf = lane & 15;
  const bool hi   = lane >= 16;
  const long row0 = (long)blockIdx.x * 16;

  const float4* X4 = reinterpret_cast<const float4*>(X + row0 * kD);
  for (int e = tid; e < 512; e += 256) {        // 16 rows x 32 float4
    const int r = e >> 5, c4 = e & 31;
    float4 xv = X4[r * 32 + c4];
    Xs[r * XS + c4 * 4 + 0] = xv.x;
    Xs[r * XS + c4 * 4 + 1] = xv.y;
    Xs[r * XS + c4 * 4 + 2] = xv.z;
    Xs[r * XS + c4 * 4 + 3] = xv.w;
  }
  __syncthreads();

  v8f aq = {0.f,0.f,0.f,0.f,0.f,0.f,0.f,0.f};
  v8f ak = {0.f,0.f,0.f,0.f,0.f,0.f,0.f,0.f};
  v8f av = {0.f,0.f,0.f,0.f,0.f,0.f,0.f,0.f};
  const int coln = 16 * wave + lhalf;           // output column of this lane

  for (int c = 0; c < 32; ++c) {                // K = 128 = 32 x 4
    const int d0 = 4 * c + (hi ? 2 : 0);
    v2f af;                                     // A frag: X[row=lhalf, d0..d0+1]
    af.x = Xs[lhalf * XS + d0];
    af.y = Xs[lhalf * XS + d0 + 1];
    const float* pq = Wq + d0 * kM + coln;
    const float* pk = Wk + d0 * kM + coln;
    const float* pv = Wv + d0 * kM + coln;
    v2f bf;
    bf.x = pq[0]; bf.y = pq[kM];                // B frag: rows d0, d0+1
    aq = wmma_f32x4(af, bf, aq);
    bf.x = pk[0]; bf.y = pk[kM];
    ak = wmma_f32x4(af, bf, ak);
    bf.x = pv[0]; bf.y = pv[kM];
    av = wmma_f32x4(af, bf, av);
  }

  const float biq = bq[coln], bik = bk[coln], biv = bv[coln];
  for (int r = 0; r < 8; ++r) {
    const long grow = row0 + (hi ? 8 + r : r);
    qo[grow * kM + coln] = aq[r] + biq;
    ko[grow * kM + coln] = ak[r] + bik;
    vo[grow * kM + coln] = av[r] + biv;
  }
}

// ---------------------------------------------------------------------------
// Kernel 2: attention for one (b, a, head).
//   Q[30x16]  x  Kall[480x16]^T  -> scores (WMMA, 60 tiles over 8 waves)
//   + geom_bias / mask(-inf), softmax over 480, attn streamed to global,
//   ctx = P[30x480] @ Vall[480x16] (split-K over waves, LDS float atomics).
// ---------------------------------------------------------------------------
__global__ __launch_bounds__(256) void attn_kernel(
    const float* __restrict__ q, const float* __restrict__ kf,
    const float* __restrict__ vf, const float* __restrict__ geom_bias,
    const int* __restrict__ nidx, const unsigned char* __restrict__ nmask,
    float* __restrict__ attn, float* __restrict__ ctx_ws) {
  constexpr int KS = 17;        // Q/K/V LDS row stride
  constexpr int SP = 481;       // score LDS row stride
  __shared__ float Qs[32 * KS];
  __shared__ float Kall[kCL * KS];
  __shared__ float Vall[kCL * KS];
  __shared__ float S[32 * SP];
  __shared__ float cb[kCL];     // bias-or-(-inf) per key column
  __shared__ float ctxAcc[32 * kHD];

  const int wg   = blockIdx.x;          // (b*A + a)*H + h
  const int h    = wg % kH;
  const int ba   = wg / kH;
  const int b    = ba / kA;
  const int tid  = threadIdx.x;
  const int wave = tid >> 5;
  const int lane = tid & 31;
  const int lhalf = lane & 15;
  const bool hi  = lane >= 16;
  const float NEG_INF = -__builtin_inff();

  // ---- stage 1: gather Q / K / V / bias into LDS -------------------------
  const float* qbase = q + (size_t)ba * kL * kM + h * kHD;
  for (int e = tid; e < 32 * kHD; e += 256) {
    const int l = e >> 4, d = e & 15;
    Qs[l * KS + d] = (l < kL) ? qbase[l * kM + d] : 0.0f;
  }
  for (int kk = 0; kk < kK; ++kk) {
    const int j = nidx[ba * kK + kk];
    const float* kb = kf + ((size_t)(b * kA + j) * kL) * kM + h * kHD;
    const float* vb = vf + ((size_t)(b * kA + j) * kL) * kM + h * kHD;
    for (int e = tid; e < kL * kHD; e += 256) {
      const int l = e >> 4, d = e & 15;
      Kall[(kk * kL + l) * KS + d] = kb[l * kM + d];
      Vall[(kk * kL + l) * KS + d] = vb[l * kM + d];
    }
  }
  for (int col = tid; col < kCL; col += 256) {
    const int kk = col / kL;
    const bool m = nmask[ba * kK + kk] != 0;
    cb[col] = m ? geom_bias[ba * kK + kk] : NEG_INF;
  }
  __syncthreads();

  // ---- stage 2: scores = Q @ K^T * scale + bias --------------------------
  for (int t = wave; t < 60; t += 8) {          // 2 M-tiles x 30 N-tiles
    const int mt = t & 1, nt = t >> 1;
    v8f acc = {0.f,0.f,0.f,0.f,0.f,0.f,0.f,0.f};
    const int qrow = mt * 16 + lhalf;
    const int krow = nt * 16 + lhalf;
    for (int c = 0; c < 4; ++c) {               // inner dim 16 = 4 x 4
      const int d0 = 4 * c + (hi ? 2 : 0);
      v2f af;  af.x = Qs[qrow * KS + d0];   af.y = Qs[qrow * KS + d0 + 1];
      v2f bf;  bf.x = Kall[krow * KS + d0]; bf.y = Kall[krow * KS + d0 + 1];
      acc = wmma_f32x4(af, bf, acc);
    }
    const int colg = nt * 16 + lhalf;
    const float cbv = cb[colg];
    for (int r = 0; r < 8; ++r) {
      const int row = mt * 16 + (hi ? 8 + r : r);
      S[row * SP + colg] = (row < kL) ? (acc[r] * kSCALE + cbv) : 0.0f;
    }
  }
  __syncthreads();

  // ---- stage 3: softmax over 480 columns (8 lanes per row) ---------------
  if (tid < kL * 8) {
    const int row = tid >> 3, sub = tid & 7;
    float mx = NEG_INF;
    for (int col = sub; col < kCL; col += 8) mx = fmaxf(mx, S[row * SP + col]);
    for (int off = 4; off; off >>= 1) mx = fmaxf(mx, __shfl_xor(mx, off, 8));
    float sum = 0.0f;
    for (int col = sub; col < kCL; col += 8) {
      const float s = S[row * SP + col];
      sum += (mx == NEG_INF) ? 0.0f : __expf(s - mx);
    }
    for (int off = 4; off; off >>= 1) sum += __shfl_xor(sum, off, 8);
    const float inv = (sum > 0.0f) ? 1.0f / sum : 0.0f;   // nan_to_num rows
    float* arow = attn + (((size_t)ba * kH + h) * kL + row) * kCL;
    for (int col = sub; col < kCL; col += 8) {
      const float s = S[row * SP + col];
      const float e = (mx == NEG_INF) ? 0.0f : __expf(s - mx);
      const float p = e * inv;
      S[row * SP + col] = p;     // keep normalized probs for ctx GEMM
      arow[col] = p;             // stream attn output (dominant HBM traffic)
    }
  }
  __syncthreads();

  // ---- stage 4: ctx = P @ V, split-K across the 8 waves ------------------
  for (int e = tid; e < 32 * kHD; e += 256) ctxAcc[e] = 0.0f;
  __syncthreads();
  const int k0w = wave * 60;                    // 480 / 8 waves
  for (int mt = 0; mt < 2; ++mt) {
    v8f acc = {0.f,0.f,0.f,0.f,0.f,0.f,0.f,0.f};
    const int prow = mt * 16 + lhalf;
    for (int c = 0; c < 15; ++c) {
      const int kk0 = k0w + 4 * c + (hi ? 2 : 0);
      v2f af;  af.x = S[prow * SP + kk0];       af.y = S[prow * SP + kk0 + 1];
      v2f bf;  bf.x = Vall[kk0 * KS + lhalf];   bf.y = Vall[(kk0 + 1) * KS + lhalf];
      acc = wmma_f32x4(af, bf, acc);
    }
    for (int r = 0; r < 8; ++r) {
      const int row = mt * 16 + (hi ? 8 + r : r);
      atomicAdd(&ctxAcc[row * kHD + lhalf], acc[r]);   // ds_add_f32
    }
  }
  __syncthreads();
  float* cbase = ctx_ws + (size_t)ba * kL * kM + h * kHD;
  for (int e = tid; e < kL * kHD; e += 256) {
    const int l = e >> 4, d = e & 15;
    cbase[l * kM + d] = ctxAcc[l * kHD + d];
  }
}

// ---------------------------------------------------------------------------
// Kernel 3: output projection  ctx[15360,128] @ Wo[128,128] + bo
// ---------------------------------------------------------------------------
__global__ __launch_bounds__(256) void out_proj_kernel(
    const float* __restrict__ X, const float* __restrict__ Wo,
    const float* __restrict__ bo, float* __restrict__ out) {
  constexpr int XS = 132;
  __shared__ float Xs[16 * XS];
  const int tid   = threadIdx.x;
  const int wave  = tid >> 5;
  const int lane  = tid & 31;
  const int lhalf = lane & 15;
  const bool hi   = lane >= 16;
  const long row0 = (long)blockIdx.x * 16;

  const float4* X4 = reinterpret_cast<const float4*>(X + row0 * kM);
  for (int e = tid; e < 512; e += 256) {
    const int r = e >> 5, c4 = e & 31;
    float4 xv = X4[r * 32 + c4];
    Xs[r * XS + c4 * 4 + 0] = xv.x;
    Xs[r * XS + c4 * 4 + 1] = xv.y;
    Xs[r * XS + c4 * 4 + 2] = xv.z;
    Xs[r * XS + c4 * 4 + 3] = xv.w;
  }
  __syncthreads();

  v8f acc = {0.f,0.f,0.f,0.f,0.f,0.f,0.f,0.f};
  const int coln = 16 * wave + lhalf;
  for (int c = 0; c < 32; ++c) {
    const int d0 = 4 * c + (hi ? 2 : 0);
    v2f af;
    af.x = Xs[lhalf * XS + d0];
    af.y = Xs[lhalf * XS + d0 + 1];
    const float* pw = Wo + d0 * kM + coln;
    v2f bf; bf.x = pw[0]; bf.y = pw[kM];
    acc = wmma_f32x4(af, bf, acc);
  }
  const float bi = bo[coln];
  for (int r = 0; r < 8; ++r) {
    const long grow = row0 + (hi ? 8 + r : r);
    out[grow * kM + coln] = acc[r] + bi;
  }
}

// ---------------------------------------------------------------------------
extern "C" void kernel_launch(void* const* d_in, const int* in_sizes, int n_in,
                              void* d_out, int out_size, void* d_ws,
                              size_t ws_size, hipStream_t stream) {
  const float* future_feat = (const float*)d_in[0];
  const float* geom_bias   = (const float*)d_in[1];
  const float* Wq          = (const float*)d_in[2];
  const float* bq          = (const float*)d_in[3];
  const float* Wk          = (const float*)d_in[4];
  const float* bk          = (const float*)d_in[5];
  const float* Wv          = (const float*)d_in[6];
  const float* bv          = (const float*)d_in[7];
  const float* Wo          = (const float*)d_in[8];
  const float* bo          = (const float*)d_in[9];
  const int*   nidx        = (const int*)d_in[10];
  const unsigned char* nmask = (const unsigned char*)d_in[11];  // jax bool = 1B

  float* outp  = (float*)d_out;                  // [B,A,L,M]
  float* attnp = outp + (size_t)kR * kM;         // [B,A,H,L,K,L]

  float* ws = (float*)d_ws;
  const size_t seg = (size_t)kR * kM;            // 1,966,080 floats
  float* qw = ws;
  float* kw = ws + seg;
  float* vw = ws + 2 * seg;
  float* cw = ws + 3 * seg;

  // 1) project q/k/v once per source row (gather trick: K/V of a neighbour
  //    depend only on the source agent -> 16x fewer projection FLOPs)
  qkv_proj_kernel<<<kR / 16, 256, 0, stream>>>(future_feat, Wq, bq, Wk, bk,
                                               Wv, bv, qw, kw, vw);
  // 2) attention per (b, a, head)
  attn_kernel<<<kB * kA * kH, 256, 0, stream>>>(qw, kw, vw, geom_bias, nidx,
                                                nmask, attnp, cw);
  // 3) output projection
  out_proj_kernel<<<kR / 16, 256, 0, stream>>>(cw, Wo, bo, outp);
}